// CRF_33852932227637
// MI455X (gfx1250) — compile-verified
//
#include <hip/hip_runtime.h>

typedef __attribute__((ext_vector_type(2))) float v2f;
typedef __attribute__((ext_vector_type(4))) float v4f;
typedef __attribute__((ext_vector_type(8))) float v8f;

#define NEG_INF (-10000.0f)
#define L2E 1.4426950408889634f
#define LN2 0.6931471805599453f

constexpr int B = 512;
constexpr int T = 1024;
constexpr int N = 64;
constexpr int PAD = 68;   // row stride 68 dwords -> conflict-free WMMA A-operand loads

__device__ __forceinline__ v8f wmma_f32_16x16x4(v2f a, v2f b, v8f c) {
#if __has_builtin(__builtin_amdgcn_wmma_f32_16x16x4_f32)
  // (neg_a, A, neg_b, B, c_mod, C, reuse_a, reuse_b)
  return __builtin_amdgcn_wmma_f32_16x16x4_f32(false, a, false, b, (short)0, c,
                                               false, false);
#else
  asm volatile("v_wmma_f32_16x16x4_f32 %0, %1, %2, %0"
               : "+v"(c)
               : "v"(a), "v"(b));
  return c;
#endif
}

// exp(x): single v_mul + v_exp_f32.
__device__ __forceinline__ float fast_exp(float x) {
#if __has_builtin(__builtin_amdgcn_exp2f)
  return __builtin_amdgcn_exp2f(x * L2E);
#else
  return __expf(x);
#endif
}

// ln(x) for x guaranteed normal: raw v_log_f32 (log2) * ln2, no denorm fixup.
__device__ __forceinline__ float fast_log(float x) {
#if __has_builtin(__builtin_amdgcn_logf)
  return __builtin_amdgcn_logf(x) * LN2;
#else
  return __logf(x);
#endif
}

__global__ __launch_bounds__(128) void crf_forward_kernel(
    const float* __restrict__ h, const float* __restrict__ trans,
    const int* __restrict__ lengths, float* __restrict__ out) {
  __shared__ float s_score[16][PAD];            // absolute log-domain scores
  __shared__ float s_A[16][PAD];                // exp(score - m_stale): WMMA A
  __shared__ __align__(16) float s_pm[8][16];   // partial per-batch maxes (this step)
  __shared__ __align__(16) float s_m[2][16];    // per-batch normalizer, double-buffered

  const int tid = threadIdx.x;
  const int wave = tid >> 5;       // 0..3: state-column tile
  const int lane = tid & 31;
  const int hi = lane >> 4;        // 0/1: half split of K (A/B) and M (C)
  const int lm = lane & 15;        // A/C row (batch); B/C column
  const int col = wave * 16 + lm;  // state column this lane owns in C layout
  const int b0 = blockIdx.x * 16;  // batch tile base

  // ---- B operand: Et[j][i] = exp(trans[i,j]) in native WMMA B layout.
  // B 4x16 f32: VGPR v, lane: K = 4*kstep + v + 2*hi, N = lm.
  v2f bmat[16];
#pragma unroll
  for (int k = 0; k < 16; ++k) {
    const int K0 = 4 * k + 2 * hi;
    const float* tp = trans + (size_t)col * N + K0;  // trans[col, K0..K0+1]
    bmat[k].x = fast_exp(tp[0]);
    bmat[k].y = fast_exp(tp[1]);
  }

  // ---- Per-lane state for its 8 C-tile elements (batches r+8*hi, state=col).
  const float* hptr[8];
  int lenv[8];
  float prev[8];
#pragma unroll
  for (int r = 0; r < 8; ++r) {
    const int b = r + 8 * hi;
    hptr[r] = h + ((size_t)(b0 + b) * T) * N + col;
    lenv[r] = lengths[b0 + b];
    prev[r] = (col == N - 2) ? 0.0f : NEG_INF;  // START_TAG = N-2
  }

  // ---- Init score slab + initial normalizer (row max of init scores is 0).
  const int bb = tid & 15;   // batch this thread handles in phase X
  const int g = tid >> 4;    // 0..7: j-slice
  const int j0 = g * 8;
#pragma unroll
  for (int jj = 0; jj < 8; ++jj)
    s_score[bb][j0 + jj] = ((j0 + jj) == N - 2) ? 0.0f : NEG_INF;
  if (tid < 16) s_m[0][tid] = 0.0f;
  __syncthreads();

  for (int t = 0; t < T; ++t) {
    const int cur = t & 1;

    // Emissions for this lane's C elements (coalesced per half-wave); issue early.
    float emit[8];
#pragma unroll
    for (int r = 0; r < 8; ++r) emit[r] = hptr[r][(size_t)t * N];
    if (t + 1 < T) __builtin_prefetch(hptr[0] + (size_t)(t + 1) * N, 0, 0);

    // ---- Phase X: A = exp(score - m_stale) (one-step-stale normalizer is
    // numerically safe: per-step score growth <= ~12), plus partial max of the
    // current scores for next step's normalizer.
    const v4f sa = *(const v4f*)&s_score[bb][j0];
    const v4f sb = *(const v4f*)&s_score[bb][j0 + 4];
    const float mst = s_m[cur][bb];
    v4f ea, eb;
    ea.x = fast_exp(sa.x - mst); ea.y = fast_exp(sa.y - mst);
    ea.z = fast_exp(sa.z - mst); ea.w = fast_exp(sa.w - mst);
    eb.x = fast_exp(sb.x - mst); eb.y = fast_exp(sb.y - mst);
    eb.z = fast_exp(sb.z - mst); eb.w = fast_exp(sb.w - mst);
    *(v4f*)&s_A[bb][j0] = ea;
    *(v4f*)&s_A[bb][j0 + 4] = eb;
    const float pmax = fmaxf(fmaxf(fmaxf(sa.x, sa.y), fmaxf(sa.z, sa.w)),
                             fmaxf(fmaxf(sb.x, sb.y), fmaxf(sb.z, sb.w)));
    s_pm[g][bb] = pmax;
    __syncthreads();

    // ---- Phase Y: P(16x16) = A(16x64) x Et(64x16), 16 chained fp32 WMMAs.
    v8f c = {};
#pragma unroll
    for (int k = 0; k < 16; ++k) {
      const int K0 = 4 * k + 2 * hi;  // A 16x4 layout: lane lm = M, K = K0..K0+1
      const v2f a = *(const v2f*)&s_A[lm][K0];
      c = wmma_f32_16x16x4(a, bmat[k], c);
    }

    // Fold partial maxes into next step's normalizer (other buffer).
    if (tid < 16) {
      float mn = s_pm[0][tid];
#pragma unroll
      for (int gg = 1; gg < 8; ++gg) mn = fmaxf(mn, s_pm[gg][tid]);
      s_m[cur ^ 1][tid] = mn;
    }

    // score'[b,i] = emit + m_stale[b] + ln(P[b,i]); masked rows keep old value.
    const v4f m0 = *(const v4f*)&s_m[cur][8 * hi];
    const v4f m1 = *(const v4f*)&s_m[cur][8 * hi + 4];
    const float mv[8] = {m0.x, m0.y, m0.z, m0.w, m1.x, m1.y, m1.z, m1.w};
#pragma unroll
    for (int r = 0; r < 8; ++r) {
      float ns = emit[r] + mv[r] + fast_log(c[r]);
      ns = (t < lenv[r]) ? ns : prev[r];
      prev[r] = ns;
      s_score[r + 8 * hi][col] = ns;  // conflict-free: banks split by lane half
    }
    __syncthreads();
  }

  // Terminal: out[b] = logsumexp_i(score[b,i] + trans[END, i]), END = N-1.
  if (tid < 16) {
    const int b = tid;
    float m2 = -3.0e38f;
#pragma unroll 4
    for (int i = 0; i < N; ++i)
      m2 = fmaxf(m2, s_score[b][i] + trans[(N - 1) * N + i]);
    float s = 0.0f;
#pragma unroll 4
    for (int i = 0; i < N; ++i)
      s += __expf(s_score[b][i] + trans[(N - 1) * N + i] - m2);
    out[b0 + b] = m2 + __logf(s);
  }
}

extern "C" void kernel_launch(void* const* d_in, const int* in_sizes, int n_in,
                              void* d_out, int out_size, void* d_ws,
                              size_t ws_size, hipStream_t stream) {
  (void)in_sizes; (void)n_in; (void)d_ws; (void)ws_size; (void)out_size;
  const float* h = (const float*)d_in[0];       // [B, T, N] fp32
  const float* trans = (const float*)d_in[1];   // [N, N] fp32
  const int* lengths = (const int*)d_in[2];     // [B] int32
  float* out = (float*)d_out;                   // [B] fp32
  crf_forward_kernel<<<dim3(B / 16), dim3(128), 0, stream>>>(h, trans, lengths,
                                                             out);
}